// BuiltU_15762529977045
// MI455X (gfx1250) — compile-verified
//
#include <hip/hip_runtime.h>

// ---------------------------------------------------------------------------
// Apply 2x2 complex gate G to qubit 5 of a 13-qubit state, batch 2048.
//   out[b,l,c,r] = sum_a state[b,l,a,r] * G[a,c],  L=2^5=32, R=2^7=128
// Complex mul + qubit-sum == real 4x4 matrix Mhat on (sr0,si0,sr1,si1):
//   row0: ( gr00,-gi00, gr10,-gi10)   -> out real, c=0
//   row1: ( gi00, gr00, gi10, gr10)   -> out imag, c=0
//   row2: ( gr01,-gi01, gr11,-gi11)   -> out real, c=1
//   row3: ( gi01, gr01, gi11, gr11)   -> out imag, c=1
// One v_wmma_f32_16x16x4_f32 handles 16 r-sites: A = Mhat replicated to 16
// rows, B = 4x16 state tile, D rows 0..3 (== rows 8..11) are the outputs.
// Memory-bound (0.88 FLOP/B, ~268MB @ 23.3TB/s): all loads/stores are b32
// coalesced, every lane active, EXEC uniform (WMMA-legal).
// ---------------------------------------------------------------------------

typedef __attribute__((ext_vector_type(2))) float v2f;
typedef __attribute__((ext_vector_type(8))) float v8f;

#define NUM_QUBITS 13
#define NQ (1 << NUM_QUBITS)                     // 8192
#define BATCH 2048
#define QUBIT 5
#define LDIM (1 << QUBIT)                        // 32
#define RDIM (1 << (NUM_QUBITS - QUBIT - 1))     // 128
#define TILE_R 16
#define TILES_PER_ROW (RDIM / TILE_R)            // 8
#define TOTAL_TILES (BATCH * LDIM * TILES_PER_ROW) // 524288
#define TILES_PER_WAVE 8
#define THREADS_PER_BLOCK 256
#define NUM_WAVES (TOTAL_TILES / TILES_PER_WAVE)   // 65536
#define NUM_BLOCKS (NUM_WAVES * 32 / THREADS_PER_BLOCK) // 8192

__global__ __launch_bounds__(THREADS_PER_BLOCK)
void qgate_wmma_f32(const float* __restrict__ sre,
                    const float* __restrict__ sim,
                    const float* __restrict__ gre,
                    const float* __restrict__ gim,
                    float* __restrict__ ore,
                    float* __restrict__ oim) {
  const int tid  = blockIdx.x * THREADS_PER_BLOCK + threadIdx.x;
  const int wave = tid >> 5;
  const int lane = threadIdx.x & 31;
  const int half = lane >> 4;   // 0: a=0 components, 1: a=1 components
  const int col  = lane & 15;   // site index within 16-wide tile
  const int m4   = lane & 3;    // Mhat row (A rows replicate mod 4)
  const int c    = m4 >> 1;     // gate output column
  const int p    = m4 & 1;      // 0 = real-part row, 1 = imag-part row

  // --- Build A(16x4): lane holds A(row=lane%16, K={2*half, 2*half+1}). ---
  // Coefficients of (sr_a, si_a) for a = half, output component m4:
  //   k even (sr): p ? gi[a][c] :  gr[a][c]
  //   k odd  (si): p ? gr[a][c] : -gi[a][c]
  const float grv = gre[2 * half + c];
  const float giv = gim[2 * half + c];
  v2f amat;
  amat.x = p ? giv : grv;
  amat.y = p ? grv : -giv;

  const long tile0 = (long)wave * TILES_PER_WAVE;

#pragma unroll
  for (int i = 0; i < TILES_PER_WAVE; ++i) {
    const long tile = tile0 + i;
    const long bl   = tile >> 3;                        // / TILES_PER_ROW
    const int rblk  = (int)(tile & (TILES_PER_ROW - 1));
    // flattened (B,N) index of state[b][l][a=0][r0]
    const long base = bl * (2 * RDIM) + (long)rblk * TILE_R;
    const long addr = base + (long)half * RDIM + col;

    // --- B(4x16): rows (2*half, 2*half+1) on this half-wave, col = site ---
    v2f bmat;
    bmat.x = sre[addr];
    bmat.y = sim[addr];

    // prefetch next r-block of this stream (speculative; OOB is dropped)
    __builtin_prefetch(&sre[addr + TILE_R], 0, 0);
    __builtin_prefetch(&sim[addr + TILE_R], 0, 0);

    v8f cacc = {};
    // (neg_a, A, neg_b, B, c_mod, C, reuse_a, reuse_b)
    v8f d = __builtin_amdgcn_wmma_f32_16x16x4_f32(
        false, amat, false, bmat, (short)0, cacc, false, false);

    // A rows repeat mod 4 => rows 8..11 == rows 0..3, so every lane's
    // d[0..3] holds the 4 outputs for site `col`. Lane stores its half's
    // (real, imag) pair at the same coalesced address it loaded from.
    const float vre = half ? d[2] : d[0];
    const float vim = half ? d[3] : d[1];
    ore[addr] = vre;
    oim[addr] = vim;
  }
}

extern "C" void kernel_launch(void* const* d_in, const int* in_sizes, int n_in,
                              void* d_out, int out_size, void* d_ws, size_t ws_size,
                              hipStream_t stream) {
  (void)in_sizes; (void)n_in; (void)out_size; (void)d_ws; (void)ws_size;
  const float* state_real = (const float*)d_in[0];  // (2048, 8192)
  const float* state_imag = (const float*)d_in[1];  // (2048, 8192)
  const float* gate_real  = (const float*)d_in[2];  // (2, 2)
  const float* gate_imag  = (const float*)d_in[3];  // (2, 2)
  float* out_real = (float*)d_out;                  // (2, B, N): real plane
  float* out_imag = out_real + (long)BATCH * NQ;    // imag plane

  qgate_wmma_f32<<<NUM_BLOCKS, THREADS_PER_BLOCK, 0, stream>>>(
      state_real, state_imag, gate_real, gate_imag, out_real, out_imag);
}